// APPNPModel_8169027797596
// MI455X (gfx1250) — compile-verified
//
#include <hip/hip_runtime.h>
#include <math.h>

typedef __attribute__((ext_vector_type(2))) float v2f;
typedef __attribute__((ext_vector_type(8))) float v8f;

#define CIN   256
#define COUT  64
#define KITER 50
#define ALPHA 0.1f

// ---------------- CSR build ----------------

__global__ void init_count_kernel(int* __restrict__ cnt, int n) {
  int i = blockIdx.x * blockDim.x + threadIdx.x;
  if (i < n) cnt[i] = 0;
}

__global__ void count_kernel(const int* __restrict__ dst, int* __restrict__ cnt, int e) {
  int i = blockIdx.x * blockDim.x + threadIdx.x;
  if (i < e) atomicAdd(&cnt[dst[i]], 1);
}

// Single-block exclusive prefix sum over cnt[0..n) -> row_start[0..n]
__global__ void scan_kernel(const int* __restrict__ cnt, int* __restrict__ row_start, int n) {
  __shared__ int sdata[1024];
  __shared__ int s_total;
  int tid = threadIdx.x;
  if (tid == 0) s_total = 0;
  __syncthreads();
  for (int base = 0; base < n; base += 1024) {
    int i = base + tid;
    int v = (i < n) ? cnt[i] : 0;
    sdata[tid] = v;
    __syncthreads();
    // Hillis-Steele inclusive scan
    for (int off = 1; off < 1024; off <<= 1) {
      int t = (tid >= off) ? sdata[tid - off] : 0;
      __syncthreads();
      sdata[tid] += t;
      __syncthreads();
    }
    int incl = sdata[tid];
    int excl = incl - v;
    if (i < n) row_start[i] = s_total + excl;
    __syncthreads();
    if (tid == 1023) s_total += sdata[1023];  // chunk total
    __syncthreads();
  }
  if (threadIdx.x == 0) row_start[n] = s_total;
}

__global__ void dinv_cursor_kernel(const int* __restrict__ cnt, float* __restrict__ dinv,
                                   int* __restrict__ cursor, int n) {
  int i = blockIdx.x * blockDim.x + threadIdx.x;
  if (i < n) {
    // degree includes the self-loop: 1 + in-degree
    dinv[i] = rsqrtf(1.0f + (float)cnt[i]);
    cursor[i] = 0;
  }
}

// Packed CSR record: .x = source node, .y = bit-cast fp32 edge weight.
// One 8-byte broadcast load per edge in the hot loop instead of two 4-byte
// loads from two arrays.
__global__ void fill_kernel(const int* __restrict__ src, const int* __restrict__ dst,
                            const float* __restrict__ dinv, const int* __restrict__ row_start,
                            int* __restrict__ cursor, int2* __restrict__ csr, int e) {
  int i = blockIdx.x * blockDim.x + threadIdx.x;
  if (i >= e) return;
  int s = src[i];
  int d = dst[i];
  int pos = atomicAdd(&cursor[d], 1);
  int slot = row_start[d] + pos;
  // fold the (1-alpha) teleport factor into the edge weight
  float w = (1.0f - ALPHA) * dinv[s] * dinv[d];
  int2 rec;
  rec.x = s;
  rec.y = __float_as_int(w);
  csr[slot] = rec;
}

// ---------------- GEMM + ReLU via fp32 WMMA ----------------
// One wave computes a 16x64 output tile: 4 x v8f accumulators,
// K-loop of 256/4 = 64 V_WMMA_F32_16X16X4_F32 steps per N-tile.
__global__ void gemm_relu_kernel(const float* __restrict__ x, const float* __restrict__ W,
                                 const float* __restrict__ bias, float* __restrict__ h,
                                 int n_tiles) {
  int wave = (blockIdx.x * blockDim.x + threadIdx.x) >> 5;
  if (wave >= n_tiles) return;
  int lane = threadIdx.x & 31;
  int half = lane >> 4;   // 0: K pair {0,1}, 1: K pair {2,3} (ISA A-matrix 16x4 layout)
  int lm   = lane & 15;   // M (for A) / N (for B,C,D)
  int m0   = wave * 16;
  int row  = m0 + lm;

  v8f acc[4];
  #pragma unroll
  for (int nt = 0; nt < 4; ++nt) acc[nt] = (v8f){0.f,0.f,0.f,0.f,0.f,0.f,0.f,0.f};

  for (int k0 = 0; k0 < CIN; k0 += 4) {
    int ka = k0 + 2 * half;
    v2f a;
    a.x = x[row * CIN + ka];
    a.y = x[row * CIN + ka + 1];
    #pragma unroll
    for (int nt = 0; nt < 4; ++nt) {
      int col = nt * 16 + lm;
      v2f bb;
      bb.x = W[ka * COUT + col];
      bb.y = W[(ka + 1) * COUT + col];
      acc[nt] = __builtin_amdgcn_wmma_f32_16x16x4_f32(
          /*neg_a=*/false, a, /*neg_b=*/false, bb,
          /*c_mod=*/(short)0, acc[nt], /*reuse_a=*/false, /*reuse_b=*/false);
    }
  }

  // C/D layout: VGPR r -> row m0+r (lanes 0-15) / m0+r+8 (lanes 16-31), N = lm
  #pragma unroll
  for (int nt = 0; nt < 4; ++nt) {
    int col = nt * 16 + lm;
    float bv = bias[col];
    #pragma unroll
    for (int r = 0; r < 8; ++r) {
      float v = acc[nt][r] + bv;
      h[(m0 + r + 8 * half) * COUT + col] = v > 0.f ? v : 0.f;
    }
  }
}

// ---------------- APPNP propagation (gather, no atomics) ----------------
// One wave per node; each lane owns 2 channels (float2). A wave's 32 lanes
// read a full 256B row of hin coalesced per gathered edge. Per edge: one
// broadcast global_load_b64 (packed src+weight) + one b64 gather + 2 FMA.
// Everything is L2-resident (h = 12.8MB << 192MB L2).
__global__ void propagate_kernel(const float* __restrict__ hin, float* __restrict__ hout,
                                 const float* __restrict__ h0, const float* __restrict__ dinv,
                                 const int* __restrict__ row_start,
                                 const int2* __restrict__ csr, int n) {
  int node = (blockIdx.x * blockDim.x + threadIdx.x) >> 5;
  if (node >= n) return;
  int lane = threadIdx.x & 31;
  int c2 = (node * COUT + lane * 2) >> 1;  // float2 index of this lane's channel pair

  const float2* hin2 = (const float2*)hin;
  const float2* h02  = (const float2*)h0;

  float di = dinv[node];
  float sw = (1.0f - ALPHA) * di * di;     // self-loop weight with teleport folded in
  float2 hv = hin2[c2];
  float accx = sw * hv.x;
  float accy = sw * hv.y;

  int beg = row_start[node];
  int end = row_start[node + 1];
  for (int j = beg; j < end; ++j) {
    int2  rec = csr[j];                 // uniform across wave -> broadcast load
    float w   = __int_as_float(rec.y);
    float2 v  = hin2[rec.x * (COUT / 2) + lane];
    accx += w * v.x;
    accy += w * v.y;
  }

  float2 x0 = h02[c2];
  float2 o;
  o.x = accx + ALPHA * x0.x;
  o.y = accy + ALPHA * x0.y;
  ((float2*)hout)[c2] = o;
}

// ---------------- launch ----------------

extern "C" void kernel_launch(void* const* d_in, const int* in_sizes, int n_in,
                              void* d_out, int out_size, void* d_ws, size_t ws_size,
                              hipStream_t stream) {
  const float* x    = (const float*)d_in[0];
  const int*   ei   = (const int*)d_in[1];
  const float* W    = (const float*)d_in[2];
  const float* bias = (const float*)d_in[3];

  const int N = in_sizes[0] / CIN;   // 50000
  const int E = in_sizes[1] / 2;     // 800000
  const int* src = ei;               // edge_index[0]
  const int* dst = ei + E;           // edge_index[1]

  char* ws = (char*)d_ws;
  size_t off = 0;
  auto alloc = [&](size_t bytes) -> void* {
    void* p = ws + off;
    off += (bytes + 255) & ~(size_t)255;
    return p;
  };

  float* h0   = (float*)alloc((size_t)N * COUT * sizeof(float)); // x0 = relu(xW+b)
  float* h1   = (float*)alloc((size_t)N * COUT * sizeof(float)); // ping buffer
  float* dinv = (float*)alloc((size_t)N * sizeof(float));
  int*   cnt  = (int*)  alloc((size_t)N * sizeof(int));
  int*   rs   = (int*)  alloc((size_t)(N + 1) * sizeof(int));
  int*   cur  = (int*)  alloc((size_t)N * sizeof(int));
  int2*  csr  = (int2*) alloc((size_t)E * sizeof(int2));

  // CSR build (one-time cost)
  init_count_kernel<<<(N + 255) / 256, 256, 0, stream>>>(cnt, N);
  count_kernel<<<(E + 255) / 256, 256, 0, stream>>>(dst, cnt, E);
  scan_kernel<<<1, 1024, 0, stream>>>(cnt, rs, N);
  dinv_cursor_kernel<<<(N + 255) / 256, 256, 0, stream>>>(cnt, dinv, cur, N);
  fill_kernel<<<(E + 255) / 256, 256, 0, stream>>>(src, dst, dinv, rs, cur, csr, E);

  // h0 = relu(x @ W + b) via fp32 WMMA (N = 50000 = 3125 * 16)
  int n_tiles = N / 16;
  int gemm_threads = n_tiles * 32;
  gemm_relu_kernel<<<(gemm_threads + 255) / 256, 256, 0, stream>>>(x, W, bias, h0, n_tiles);

  // 50 propagation steps; even iterations write d_out so iter 50 ends there.
  float* dout = (float*)d_out;
  const float* hin = h0;
  float* hout = h1;
  int prop_blocks = (N * 32 + 255) / 256;  // one wave per node, 8 nodes per 256-thread block
  for (int it = 1; it <= KITER; ++it) {
    propagate_kernel<<<prop_blocks, 256, 0, stream>>>(hin, hout, h0, dinv, rs, csr, N);
    if (it & 1) { hin = h1;   hout = dout; }
    else        { hin = dout; hout = h1;   }
  }
}